// MultiViewRGCN_69312182223081
// MI455X (gfx1250) — compile-verified
//
#include <hip/hip_runtime.h>

typedef float     v2f  __attribute__((ext_vector_type(2)));
typedef float     v8f  __attribute__((ext_vector_type(8)));
typedef _Float16  v4h  __attribute__((ext_vector_type(4)));

// ---------------------------------------------------------------------------
// Unified WMMA GEMM:  C[m,n] (=|+=) sum_k A[m,k] * Bcol(n,k)  (+ bias[n])
//   Bcol(n,k) = Bm[ (n>>7)*16384 + (n&127) + k*128 ]
// covers bases [B,128,128] (n = b*128+o), loopw / fusion_w slices [128,128].
//
// One wave computes a 64x64 tile: 4 row-subtiles x 4 col-tiles = 16
// V_WMMA_F32_16X16X4_F32 accumulators (full fp32 precision — GEMM is
// compute-trivial on MI455X, ~30 GFLOP total, all operands L2-resident).
// Per k-step: 4 x b64 A-loads + 8 x b32 B-loads feed 16 WMMAs (B fragments
// shared across the 4 row-subtiles), vs 9 VMEM / 4 WMMA before.
//
// Fragment layouts per CDNA5 ISA 7.12.2:
//   A 16x4 : lane(half,lr): vgpr t holds A[m+lr, k + 2*half + t]
//   B 4x16 : lane(half,lr): vgpr t holds B[k + 2*half + t, n0+lr]
//   C 16x16: vgpr r -> M = mbase + r + 8*half, N = n0 + lr
//
// Row edge: A row index clamped to M-1 for loads (WMMA requires full EXEC,
// no predication); stores guarded per 16-row subtile (M % 16 == 0 here).
// If C16 != null the result is rounded to f16 (proj buffer) — compute and
// accumulation remain f32; only the stored value is rounded.
// ---------------------------------------------------------------------------
__global__ void __launch_bounds__(128)
wmma_gemm_k(const float* __restrict__ A, int lda,
            const float* __restrict__ Bm,
            const float* __restrict__ bias,
            float* __restrict__ C, _Float16* __restrict__ C16,
            int ldc, int M, int K, int accumulate)
{
    const int lane = threadIdx.x & 31;
    const int wave = threadIdx.x >> 5;
    const int half = lane >> 4;
    const int lr   = lane & 15;

    const int m0 = (blockIdx.y * 4 + wave) << 6;   // 64 rows per wave
    if (m0 >= M) return;
    const int n0 = blockIdx.x << 6;                // 64 columns per block

    // B column pointers (shared by all 4 row-subtiles)
    const float* bPtr[4];
#pragma unroll
    for (int j = 0; j < 4; ++j) {
        const int c = n0 + (j << 4) + lr;
        bPtr[j] = Bm + ((size_t)(c >> 7) << 14) + (c & 127)
                     + ((size_t)(half << 1) << 7);
    }

    // A row pointers for 4 row-subtiles (clamped for safe OOB loads)
    const float* aPtr[4];
#pragma unroll
    for (int s = 0; s < 4; ++s) {
        int row = m0 + (s << 4) + lr;
        row = (row < M) ? row : (M - 1);
        aPtr[s] = A + (size_t)row * lda + (half << 1);
    }

    v8f acc[4][4] = {};   // [row-subtile][col-tile]

    for (int k = 0; k < K; k += 4) {
        const size_t ko = (size_t)k << 7;          // k*128
        v2f b[4];
#pragma unroll
        for (int j = 0; j < 4; ++j) {
            b[j].x = bPtr[j][ko];                  // B[k+2h  , n]
            b[j].y = bPtr[j][ko + 128];            // B[k+2h+1, n]
        }
        v2f a[4];
#pragma unroll
        for (int s = 0; s < 4; ++s)
            a[s] = *(const v2f*)(aPtr[s] + k);     // A[m, k+2h], A[m, k+2h+1]
#pragma unroll
        for (int s = 0; s < 4; ++s)
#pragma unroll
            for (int j = 0; j < 4; ++j)
                acc[s][j] = __builtin_amdgcn_wmma_f32_16x16x4_f32(
                    false, a[s], false, b[j], (short)0, acc[s][j], false, false);
    }

#pragma unroll
    for (int s = 0; s < 4; ++s) {
        const int mbase = m0 + (s << 4);
        if (mbase >= M) break;                     // M % 16 == 0 -> whole subtile valid
#pragma unroll
        for (int j = 0; j < 4; ++j) {
            const int n = n0 + (j << 4) + lr;
            const float badd = bias ? bias[n] : 0.0f;
#pragma unroll
            for (int r = 0; r < 8; ++r) {
                const size_t idx = (size_t)(mbase + r + (half << 3)) * ldc + n;
                const float v = acc[s][j][r] + badd;
                if (C16) {
                    C16[idx] = (_Float16)v;
                } else {
                    C[idx] = accumulate ? (C[idx] + v) : v;
                }
            }
        }
    }
}

// ---------------------------------------------------------------------------
// Edge combine + scatter. One wave owns 32 edges: lanes load src/dst/rel
// coalesced once, then broadcast per edge with __shfl (no redundant VMEM).
// Per edge: msg = sum_b coef[rel,b] * proj_f16[src, b*128:(b+1)*128]
// Lane l reads 4 contiguous halfs per basis (4 x 8B loads = 1KB/edge from
// L2-resident proj), combines in fp32, then 4 native f32 atomics into agg.
// ---------------------------------------------------------------------------
__global__ void __launch_bounds__(256)
edge_scatter_k(const _Float16* __restrict__ proj,  // [N,512] f16
               const int*      __restrict__ src,
               const int*      __restrict__ dst,
               const int*      __restrict__ rel,
               const float*    __restrict__ coef,  // [R,4]
               float* __restrict__ agg,            // [N,128] f32
               int E)
{
    const int lane = threadIdx.x & 31;
    const int w    = (int)((blockIdx.x * blockDim.x + threadIdx.x) >> 5);

    const int e0 = w * 32;
    if (e0 >= E) return;

    int sv = 0, dv = 0, rv = 0;
    const int e = e0 + lane;
    if (e < E) { sv = src[e]; dv = dst[e]; rv = rel[e]; }
    const int cnt = (E - e0 < 32) ? (E - e0) : 32;

    for (int i = 0; i < cnt; ++i) {
        const int s = __shfl(sv, i);
        const int d = __shfl(dv, i);
        const int r = __shfl(rv, i);

        const _Float16* row = proj + ((size_t)s << 9);
        const float4 cf = *(const float4*)(coef + ((size_t)r << 2));

        const v4h q0 = *(const v4h*)(row +       (lane << 2));
        const v4h q1 = *(const v4h*)(row + 128 + (lane << 2));
        const v4h q2 = *(const v4h*)(row + 256 + (lane << 2));
        const v4h q3 = *(const v4h*)(row + 384 + (lane << 2));

        float4 acc;
        acc.x = fmaf(cf.x, (float)q0[0], fmaf(cf.y, (float)q1[0],
                fmaf(cf.z, (float)q2[0], cf.w * (float)q3[0])));
        acc.y = fmaf(cf.x, (float)q0[1], fmaf(cf.y, (float)q1[1],
                fmaf(cf.z, (float)q2[1], cf.w * (float)q3[1])));
        acc.z = fmaf(cf.x, (float)q0[2], fmaf(cf.y, (float)q1[2],
                fmaf(cf.z, (float)q2[2], cf.w * (float)q3[2])));
        acc.w = fmaf(cf.x, (float)q0[3], fmaf(cf.y, (float)q1[3],
                fmaf(cf.z, (float)q2[3], cf.w * (float)q3[3])));

        float* outp = agg + ((size_t)d << 7) + (lane << 2);
        unsafeAtomicAdd(outp + 0, acc.x);
        unsafeAtomicAdd(outp + 1, acc.y);
        unsafeAtomicAdd(outp + 2, acc.z);
        unsafeAtomicAdd(outp + 3, acc.w);
    }
}

__global__ void __launch_bounds__(256)
relu_k(float* __restrict__ x, int n4)
{
    const int i = blockIdx.x * blockDim.x + threadIdx.x;
    if (i < n4) {
        float4 v = ((float4*)x)[i];
        v.x = fmaxf(v.x, 0.0f);
        v.y = fmaxf(v.y, 0.0f);
        v.z = fmaxf(v.z, 0.0f);
        v.w = fmaxf(v.w, 0.0f);
        ((float4*)x)[i] = v;
    }
}

// ---------------------------------------------------------------------------
// Input layout (setup_inputs dict order):
//   0: node_features [N,128]
//   per view v=0..2, base = 1 + v*11:
//     +0 src, +1 dst, +2 rel,
//     +3 bases1, +4 coef1, +5 loopw1, +6 bias1,
//     +7 bases2, +8 coef2, +9 loopw2, +10 bias2
//   34: fusion_w [384,128], 35: fusion_b [128]
// Workspace: proj_f16 [N,512] (51.2MB) | agg1 [N,128] | agg2 [N,128] (f32)
// ---------------------------------------------------------------------------
extern "C" void kernel_launch(void* const* d_in, const int* in_sizes, int n_in,
                              void* d_out, int out_size, void* d_ws, size_t ws_size,
                              hipStream_t stream)
{
    const float* h0 = (const float*)d_in[0];
    const int N = in_sizes[0] / 128;

    _Float16* proj16 = (_Float16*)d_ws;
    float*    agg1   = (float*)((char*)d_ws + (size_t)N * 512 * sizeof(_Float16));
    float*    agg2   = agg1 + (size_t)N * 128;
    float*    out    = (float*)d_out;

    const dim3 blk(128);
    const int  rowBlocks = (N + 255) / 256;        // 4 waves * 64 rows
    const dim3 gridProj(8, rowBlocks);             // 512 cols
    const dim3 gridLoop(2, rowBlocks);             // 128 cols

    const float* fusion_w = (const float*)d_in[34];
    const float* fusion_b = (const float*)d_in[35];

    for (int v = 0; v < 3; ++v) {
        const int base = 1 + v * 11;
        const int*   src    = (const int*)  d_in[base + 0];
        const int*   dst    = (const int*)  d_in[base + 1];
        const int*   rel    = (const int*)  d_in[base + 2];
        const float* bases1 = (const float*)d_in[base + 3];
        const float* coef1  = (const float*)d_in[base + 4];
        const float* loopw1 = (const float*)d_in[base + 5];
        const float* bias1  = (const float*)d_in[base + 6];
        const float* bases2 = (const float*)d_in[base + 7];
        const float* coef2  = (const float*)d_in[base + 8];
        const float* loopw2 = (const float*)d_in[base + 9];
        const float* bias2  = (const float*)d_in[base + 10];
        const int E = in_sizes[base];
        const int edgeWaves  = (E + 31) / 32;      // one wave per 32 edges
        const int edgeBlocks = (edgeWaves + 7) / 8;
        const int n4 = N * 32;                     // N*128 floats as float4

        // Layer 1: proj = f16(h0 @ bases1) ; agg1 = h0 @ loopw1 + bias1
        wmma_gemm_k<<<gridProj, blk, 0, stream>>>(h0, 128, bases1, nullptr,
                                                  nullptr, proj16, 512, N, 128, 0);
        wmma_gemm_k<<<gridLoop, blk, 0, stream>>>(h0, 128, loopw1, bias1,
                                                  agg1, nullptr, 128, N, 128, 0);
        edge_scatter_k<<<edgeBlocks, 256, 0, stream>>>(proj16, src, dst, rel,
                                                       coef1, agg1, E);
        relu_k<<<(n4 + 255) / 256, 256, 0, stream>>>(agg1, n4);

        // Layer 2: proj = f16(h1 @ bases2) ; agg2 = h1 @ loopw2 + bias2
        wmma_gemm_k<<<gridProj, blk, 0, stream>>>(agg1, 128, bases2, nullptr,
                                                  nullptr, proj16, 512, N, 128, 0);
        wmma_gemm_k<<<gridLoop, blk, 0, stream>>>(agg1, 128, loopw2, bias2,
                                                  agg2, nullptr, 128, N, 128, 0);
        edge_scatter_k<<<edgeBlocks, 256, 0, stream>>>(proj16, src, dst, rel,
                                                       coef2, agg2, E);

        // Fusion: out (=|+=) h2_v @ fusion_w[v*128:(v+1)*128, :] (+ fusion_b)
        wmma_gemm_k<<<gridLoop, blk, 0, stream>>>(agg2, 128,
                                                  fusion_w + (size_t)v * 128 * 128,
                                                  (v == 0) ? fusion_b : nullptr,
                                                  out, nullptr, 128, N, 128,
                                                  (v == 0) ? 0 : 1);
    }
}